// GATv2_6047313953594
// MI455X (gfx1250) — compile-verified
//
#include <hip/hip_runtime.h>
#include <math.h>

#define HEADS 4
#define CH 16
#define DIM 64
#define NEG_SLOPE 0.2f
#define LN_EPS 1e-5f

typedef float v2f __attribute__((ext_vector_type(2)));
typedef float v8f __attribute__((ext_vector_type(8)));

// ---------------- utility ----------------

__global__ void fill_kernel(float* __restrict__ p, float v, long long n) {
    long long i = (long long)blockIdx.x * blockDim.x + threadIdx.x;
    long long stride = (long long)gridDim.x * blockDim.x;
    for (; i < n; i += stride) p[i] = v;
}

__device__ __forceinline__ void atomicMaxFloat(float* addr, float val) {
    if (val >= 0.0f) {
        atomicMax((int*)addr, __float_as_int(val));
    } else {
        atomicMin((unsigned int*)addr, __float_as_uint(val));
    }
}

// ---------------- dense projections: xl = X*Wl, xr = X*Wr via WMMA f32 16x16x4 ----------------
// Both 64x64 weight matrices are staged in LDS once per block (shared by all 8
// waves). Each wave owns one 16-row tile of X: A fragments are loaded once and
// reused across 4 column tiles x 2 weight matrices = 8 accumulators. For each
// output tile, all 16 B fragments are preloaded from LDS into registers so the
// 16 v_wmma ops issue back-to-back instead of waiting on memory per step.

__global__ void gemm64_wmma(const float* __restrict__ X,
                            const float* __restrict__ Wl,
                            const float* __restrict__ Wr,
                            float* __restrict__ xl,
                            float* __restrict__ xr,
                            int mtiles) {
    __shared__ float sW[2][DIM * DIM];   // 32 KB: Wl then Wr

    // cooperative copy: 256 threads x 2 matrices x 4 float4 each
    {
        const float4* gl = (const float4*)Wl;
        const float4* gr = (const float4*)Wr;
        float4* s0 = (float4*)sW[0];
        float4* s1 = (float4*)sW[1];
        const int nv = DIM * DIM / 4;    // 1024 float4 per matrix
#pragma unroll
        for (int i = threadIdx.x; i < nv; i += 256) {
            s0[i] = gl[i];
            s1[i] = gr[i];
        }
    }
    __syncthreads();

    int wave = threadIdx.x >> 5;
    int lane = threadIdx.x & 31;
    int mt = blockIdx.x * (blockDim.x >> 5) + wave;
    if (mt >= mtiles) return;            // wave-uniform exit (EXEC stays all-1s)

    int r16   = lane & 15;               // row within tile (A), col within tile (B/C)
    int khalf = (lane >> 4) * 2;         // 0 or 2: K sub-offset per ISA A-layout

    // Load all A fragments for K = 0..63 (16 steps of K=4)
    v2f a[16];
    const float* xrow = X + (size_t)(mt * 16 + r16) * DIM;
#pragma unroll
    for (int kk = 0; kk < 16; ++kk) {
        a[kk].x = xrow[kk * 4 + khalf + 0];
        a[kk].y = xrow[kk * 4 + khalf + 1];
    }

    float* Os[2] = {xl, xr};
    int rbase = mt * 16 + 8 * (lane >> 4); // C/D layout: VGPR v -> row v / v+8

#pragma unroll
    for (int m = 0; m < 2; ++m) {
        const float* W = sW[m];
        float*       O = Os[m];
#pragma unroll
        for (int nt = 0; nt < 4; ++nt) {
            int col = nt * 16 + r16;

            // preload all B fragments for this tile from LDS
            v2f b[16];
#pragma unroll
            for (int kk = 0; kk < 16; ++kk) {
                b[kk].x = W[(kk * 4 + khalf + 0) * DIM + col];
                b[kk].y = W[(kk * 4 + khalf + 1) * DIM + col];
            }

            v8f c = {};
#pragma unroll
            for (int kk = 0; kk < 16; ++kk) {
                c = __builtin_amdgcn_wmma_f32_16x16x4_f32(
                        false, a[kk], false, b[kk], (short)0, c, false, false);
            }
#pragma unroll
            for (int v = 0; v < 8; ++v)
                O[(size_t)(rbase + v) * DIM + col] = c[v];
        }
    }
}

// ---------------- edge pass 1: logits e = <leakyrelu(xl[s]+xr[d]), att>, segment max ----------------

__global__ void edge_logits(const float* __restrict__ xl, const float* __restrict__ xr,
                            const long long* __restrict__ ei, const float* __restrict__ att,
                            float* __restrict__ ebuf, float* __restrict__ emax,
                            int Etot, int Ereal) {
    __shared__ float satt[DIM];
    if (threadIdx.x < DIM) satt[threadIdx.x] = att[threadIdx.x];
    __syncthreads();
    long long idx = (long long)blockIdx.x * blockDim.x + threadIdx.x;
    if (idx >= (long long)Etot * HEADS) return;
    int e = (int)(idx >> 2), h = (int)(idx & 3);
    long long s, d;
    if (e < Ereal) { s = ei[e]; d = ei[(long long)Ereal + e]; }
    else           { s = d = e - Ereal; }
    const float* pl = xl + s * DIM + h * CH;
    const float* pr = xr + d * DIM + h * CH;
    float acc = 0.0f;
#pragma unroll
    for (int c = 0; c < CH; ++c) {
        float m = pl[c] + pr[c];
        m = (m > 0.0f) ? m : m * NEG_SLOPE;
        acc += m * satt[h * CH + c];
    }
    ebuf[idx] = acc;
    atomicMaxFloat(&emax[d * HEADS + h], acc);
}

// ---------------- edge pass 2: p = exp(e - max), segment sum ----------------

__global__ void edge_softmax(const long long* __restrict__ ei,
                             float* __restrict__ ebuf, const float* __restrict__ emax,
                             float* __restrict__ denom, int Etot, int Ereal) {
    long long idx = (long long)blockIdx.x * blockDim.x + threadIdx.x;
    if (idx >= (long long)Etot * HEADS) return;
    int e = (int)(idx >> 2), h = (int)(idx & 3);
    long long d;
    if (e < Ereal) d = ei[(long long)Ereal + e];
    else           d = e - Ereal;
    float p = expf(ebuf[idx] - emax[d * HEADS + h]);
    ebuf[idx] = p;
    atomicAdd(&denom[d * HEADS + h], p);
}

// ---------------- edge pass 3: out[d] += (p/denom) * xl[s] ----------------

__global__ void edge_aggregate(const long long* __restrict__ ei,
                               const float* __restrict__ ebuf, const float* __restrict__ denom,
                               const float* __restrict__ xl, float* __restrict__ out,
                               int Etot, int Ereal) {
    long long idx = (long long)blockIdx.x * blockDim.x + threadIdx.x;
    if (idx >= (long long)Etot * HEADS) return;
    int e = (int)(idx >> 2), h = (int)(idx & 3);
    long long s, d;
    if (e < Ereal) { s = ei[e]; d = ei[(long long)Ereal + e]; }
    else           { s = d = e - Ereal; }
    float alpha = ebuf[idx] / denom[d * HEADS + h];
    const float* pl = xl + s * DIM + h * CH;
    float*       po = out + d * DIM + h * CH;
#pragma unroll
    for (int c = 0; c < CH; ++c)
        atomicAdd(&po[c], alpha * pl[c]);
}

// ---------------- epilogue: +bias, +residual, LayerNorm, exact GELU (wave per node) ----------------

__global__ void finalize_node(const float* __restrict__ agg, const float* __restrict__ bias,
                              const float* __restrict__ xres,
                              const float* __restrict__ gamma, const float* __restrict__ beta,
                              float* __restrict__ xout, int n) {
    int node = blockIdx.x * (blockDim.x >> 5) + (threadIdx.x >> 5);
    if (node >= n) return;
    int lane = threadIdx.x & 31;
    size_t base = (size_t)node * DIM;
    float v0 = agg[base + lane]      + bias[lane]      + xres[base + lane];
    float v1 = agg[base + 32 + lane] + bias[32 + lane] + xres[base + 32 + lane];
    float s = v0 + v1, ss = v0 * v0 + v1 * v1;
#pragma unroll
    for (int off = 16; off > 0; off >>= 1) {
        s  += __shfl_xor(s,  off, 32);
        ss += __shfl_xor(ss, off, 32);
    }
    float mu  = s * (1.0f / DIM);
    float var = ss * (1.0f / DIM) - mu * mu;
    float inv = rsqrtf(var + LN_EPS);
    v0 = (v0 - mu) * inv * gamma[lane]      + beta[lane];
    v1 = (v1 - mu) * inv * gamma[32 + lane] + beta[32 + lane];
    v0 = 0.5f * v0 * (1.0f + erff(v0 * 0.70710678118654752f));
    v1 = 0.5f * v1 * (1.0f + erff(v1 * 0.70710678118654752f));
    xout[base + lane]      = v0;
    xout[base + 32 + lane] = v1;
}

// ---------------- host launcher ----------------

extern "C" void kernel_launch(void* const* d_in, const int* in_sizes, int n_in,
                              void* d_out, int out_size, void* d_ws, size_t ws_size,
                              hipStream_t stream) {
    const float*     x    = (const float*)d_in[0];
    const long long* ei   = (const long long*)d_in[1];
    const float*     Wl   = (const float*)d_in[2];   // [3,64,64]
    const float*     Wr   = (const float*)d_in[3];   // [3,64,64]
    const float*     att  = (const float*)d_in[4];   // [3,4,16]
    const float*     bias = (const float*)d_in[5];   // [3,64]
    const float*     gam  = (const float*)d_in[6];   // [3,64]
    const float*     bet  = (const float*)d_in[7];   // [3,64]
    float*           yout = (float*)d_out;           // [N,64]

    const int N    = in_sizes[0] / DIM;
    const int E    = in_sizes[1] / 2;
    const int Etot = E + N;                          // with self-loops

    // workspace carve-up (floats)
    float* ws    = (float*)d_ws;
    float* xl    = ws;
    float* xr    = xl + (size_t)N * DIM;             // also reused as aggregation accumulator
    float* xbufA = xr + (size_t)N * DIM;
    float* xbufB = xbufA + (size_t)N * DIM;
    float* ebuf  = xbufB + (size_t)N * DIM;
    float* emax  = ebuf + (size_t)Etot * HEADS;
    float* denom = emax + (size_t)N * HEADS;

    const int BLK = 256;
    const int mtiles = (N + 15) / 16;
    const int gemmBlocks = (mtiles + 7) / 8;
    const long long edgeWork = (long long)Etot * HEADS;
    const int edgeBlocks = (int)((edgeWork + BLK - 1) / BLK);
    const int nodeBlocks = (N + 7) / 8;              // 8 waves per block, wave per node
    const int fillBlocks = 2048;

    for (int i = 0; i < 3; ++i) {
        const float* xin  = (i == 0) ? x : (i == 1 ? xbufA : xbufB);
        float*       xout = (i == 0) ? xbufA : (i == 1 ? xbufB : yout);

        // projections
        gemm64_wmma<<<gemmBlocks, BLK, 0, stream>>>(
            xin, Wl + (size_t)i * DIM * DIM, Wr + (size_t)i * DIM * DIM, xl, xr, mtiles);

        // segment-max init
        fill_kernel<<<fillBlocks, BLK, 0, stream>>>(emax, -INFINITY, (long long)N * HEADS);

        // pass 1: logits + running max (reads xr for the last time)
        edge_logits<<<edgeBlocks, BLK, 0, stream>>>(
            xl, xr, ei, att + (size_t)i * HEADS * CH, ebuf, emax, Etot, E);

        // xr is now dead -> reuse as aggregation accumulator
        fill_kernel<<<fillBlocks, BLK, 0, stream>>>(xr, 0.0f, (long long)N * DIM);
        fill_kernel<<<fillBlocks, BLK, 0, stream>>>(denom, 0.0f, (long long)N * HEADS);

        // pass 2: exp + denom
        edge_softmax<<<edgeBlocks, BLK, 0, stream>>>(ei, ebuf, emax, denom, Etot, E);

        // pass 3: weighted scatter-add
        edge_aggregate<<<edgeBlocks, BLK, 0, stream>>>(ei, ebuf, denom, xl, xr, Etot, E);

        // bias + residual + LN + GELU
        finalize_node<<<nodeBlocks, BLK, 0, stream>>>(
            xr, bias + (size_t)i * DIM, xin, gam + (size_t)i * DIM, bet + (size_t)i * DIM,
            xout, N);
    }
}